// LMorph_20907900797012
// MI455X (gfx1250) — compile-verified
//
#include <hip/hip_runtime.h>

#ifndef __has_builtin
#define __has_builtin(x) 0
#endif

#define LM_N 4
#define LM_C 2
#define LM_H 256
#define LM_W 256
#define LM_O 4
#define LM_KH 5
#define LM_KW 5
#define LM_HO (LM_H - LM_KH + 1)   // 252
#define LM_WO (LM_W - LM_KW + 1)   // 252
#define TILE 16
#define ITILE (TILE + LM_KW - 1)   // 20

// ---------- CDNA5 async global->LDS path (ASYNCcnt) ----------
#if defined(__gfx1250__) && __has_builtin(__builtin_amdgcn_global_load_async_to_lds_b32)
#define LM_ASYNC_LDS 1
#else
#define LM_ASYNC_LDS 0
#endif

typedef __attribute__((address_space(1))) int lm_gint;
typedef __attribute__((address_space(3))) int lm_lint;

// ---------- fast transcendentals: single v_log_f32 / v_exp_f32 ----------
__device__ __forceinline__ float lm_log2(float x) {
#if __has_builtin(__builtin_amdgcn_logf)
  return __builtin_amdgcn_logf(x);      // v_log_f32 (log base-2)
#elif defined(__HIP_DEVICE_COMPILE__)
  return __log2f(x);
#else
  return x;  // host semantic-pass stub; never executed
#endif
}
__device__ __forceinline__ float lm_exp2(float x) {
#if __has_builtin(__builtin_amdgcn_exp2f)
  return __builtin_amdgcn_exp2f(x);     // v_exp_f32 (2^x)
#elif defined(__HIP_DEVICE_COMPILE__)
  return __exp2f(x);
#else
  return x;  // host semantic-pass stub; never executed
#endif
}

// ================= stage 1: per-block min/max partials =================
__global__ __launch_bounds__(256) void lm_minmax1(const float* __restrict__ in,
                                                  int n, float* __restrict__ ws) {
  __shared__ float smin[256];
  __shared__ float smax[256];
  const int tid = threadIdx.x;
  const int stride = gridDim.x * blockDim.x;
  float vmin = 3.402823466e38f, vmax = -3.402823466e38f;
  for (int i = blockIdx.x * blockDim.x + tid; i < n; i += stride) {
    const float v = in[i];
    vmin = fminf(vmin, v);
    vmax = fmaxf(vmax, v);
  }
  smin[tid] = vmin;
  smax[tid] = vmax;
  __syncthreads();
  for (int s = 128; s > 0; s >>= 1) {
    if (tid < s) {
      smin[tid] = fminf(smin[tid], smin[tid + s]);
      smax[tid] = fmaxf(smax[tid], smax[tid + s]);
    }
    __syncthreads();
  }
  if (tid == 0) {
    ws[blockIdx.x]       = smin[0];
    ws[256 + blockIdx.x] = smax[0];
  }
}

// ================= stage 2: final min/max -> {imin, 1/range} =================
__global__ __launch_bounds__(256) void lm_minmax2(float* __restrict__ ws) {
  __shared__ float smin[256];
  __shared__ float smax[256];
  const int tid = threadIdx.x;
  smin[tid] = ws[tid];
  smax[tid] = ws[256 + tid];
  __syncthreads();
  for (int s = 128; s > 0; s >>= 1) {
    if (tid < s) {
      smin[tid] = fminf(smin[tid], smin[tid + s]);
      smax[tid] = fmaxf(smax[tid], smax[tid + s]);
    }
    __syncthreads();
  }
  if (tid == 0) {
    const float imin = smin[0];
    const float imax = smax[0];
    ws[512] = imin;
    ws[513] = 1.0f / (imax - imin);
  }
}

// ================= main LMorph kernel =================
// grid: (ceil(Wo/16), ceil(Ho/16), N), block: (16,16) = 8 wave32 waves
__global__ __launch_bounds__(256) void lm_main(const float* __restrict__ in,
                                               const float* __restrict__ filt,
                                               const float* __restrict__ pexp,
                                               const float* __restrict__ mm,
                                               float* __restrict__ out) {
  __shared__ float xs[LM_C][ITILE][ITILE];   // raw input tile (2*20*20*4 = 3200 B)

  const int n   = blockIdx.z;
  const int tx0 = blockIdx.x * TILE;
  const int ty0 = blockIdx.y * TILE;
  const int tid = threadIdx.y * TILE + threadIdx.x;

  const float imin = mm[0];
  const float inv  = mm[1];

  // ---- cooperative halo-tile load: global -> LDS (async on CDNA5) ----
  const int tot = LM_C * ITILE * ITILE;  // 800 elements
  for (int idx = tid; idx < tot; idx += 256) {
    const int c   = idx / (ITILE * ITILE);
    const int rem = idx - c * (ITILE * ITILE);
    const int r   = rem / ITILE;
    const int col = rem - r * ITILE;
    const int gy  = min(ty0 + r,  LM_H - 1);   // clamped halo never read by active lanes
    const int gx  = min(tx0 + col, LM_W - 1);
    const float* gp = in + ((n * LM_C + c) * LM_H + gy) * LM_W + gx;
    float* lp = &xs[c][r][col];
#if LM_ASYNC_LDS
    __builtin_amdgcn_global_load_async_to_lds_b32(
        (lm_gint*)(const_cast<float*>(gp)),
        (lm_lint*)(lp),
        0, 0);
#else
    *lp = __ldg(gp);
#endif
  }
#if LM_ASYNC_LDS
#if __has_builtin(__builtin_amdgcn_s_wait_asynccnt)
  __builtin_amdgcn_s_wait_asynccnt(0);
#else
  asm volatile("s_wait_asynccnt 0" ::: "memory");
#endif
#endif
  __syncthreads();

  const int lx = threadIdx.x, ly = threadIdx.y;
  const int wo = tx0 + lx, ho = ty0 + ly;
  const bool active = (wo < LM_WO) && (ho < LM_HO);

  // rescale folded into the tap:  t = (raw*inv + base) + f,  base = 1 - imin*inv
  const float base = 1.0f - imin * inv;

  // hoist the o-invariant rescaled window into registers (50 VGPRs):
  // cuts LDS reads 4x and removes the per-tap FMA from the o-loop.
  float u[LM_C][LM_KH][LM_KW];
#pragma unroll
  for (int c = 0; c < LM_C; ++c)
#pragma unroll
    for (int i = 0; i < LM_KH; ++i)
#pragma unroll
      for (int j = 0; j < LM_KW; ++j)
        u[c][i][j] = fmaf(xs[c][ly + i][lx + j], inv, base);

#pragma unroll
  for (int o = 0; o < LM_O; ++o) {
    float num = 0.0f, den = 0.0f;
#pragma unroll
    for (int c = 0; c < LM_C; ++c) {
      const float pe = pexp[o * LM_C + c];                        // block-uniform -> s_load
      const float* fo = filt + (o * LM_C + c) * (LM_KH * LM_KW);  // block-uniform -> s_load
#pragma unroll
      for (int i = 0; i < LM_KH; ++i) {
#pragma unroll
        for (int j = 0; j < LM_KW; ++j) {
          const float t = u[c][i][j] + fo[i * LM_KW + j];  // v_add (SGPR filter)
          const float w = lm_exp2(pe * lm_log2(t));        // t^p, t >= 1
          num = fmaf(w, t, num);                           // accumulates t^(p+1)
          den += w;                                        // accumulates t^p
        }
      }
    }
    if (active) out[((n * LM_O + o) * LM_HO + ho) * LM_WO + wo] = num / den;
  }
}

// ================= launch =================
extern "C" void kernel_launch(void* const* d_in, const int* in_sizes, int n_in,
                              void* d_out, int out_size, void* d_ws, size_t ws_size,
                              hipStream_t stream) {
  const float* in   = (const float*)d_in[0];
  const float* filt = (const float*)d_in[1];
  const float* p    = (const float*)d_in[2];
  float* out = (float*)d_out;
  float* ws  = (float*)d_ws;   // [0..255] mins, [256..511] maxs, [512]=imin, [513]=1/range

  const int nelem = in_sizes[0];  // 4*2*256*256 = 524288

  lm_minmax1<<<256, 256, 0, stream>>>(in, nelem, ws);
  lm_minmax2<<<1, 256, 0, stream>>>(ws);

  dim3 grid((LM_WO + TILE - 1) / TILE, (LM_HO + TILE - 1) / TILE, LM_N);
  dim3 block(TILE, TILE);
  lm_main<<<grid, block, 0, stream>>>(in, filt, p, ws + 512, out);
}